// PaiNNInteraction_74698071212152
// MI455X (gfx1250) — compile-verified
//
#include <hip/hip_runtime.h>

#define NF 128
#define NF3 384
#define NATOMS 20000
#define NEDGES 500000

typedef __attribute__((ext_vector_type(2))) float v2f;
typedef __attribute__((ext_vector_type(8))) float v8f;

// silu via hardware v_rcp_f32 (single TRANS op, co-executes with WMMA)
// instead of IEEE div (v_div_scale + newton + v_div_fixup).
__device__ __forceinline__ float silu_f(float x) {
  return x * __builtin_amdgcn_rcpf(1.0f + __expf(-x));
}

// ---------------------------------------------------------------------------
// Kernel 1: fused node MLP  x = silu(q@W1+b1) @ W2 + b2   -> xout [N, 384]
// One block = 16 atoms, 8 waves. Wave w does h-cols [16w,16w+16) in stage 1,
// then x-col tiles {w, w+8, w+16} in stage 2. fp32 WMMA 16x16x4.
// ---------------------------------------------------------------------------
__global__ __launch_bounds__(256) void painn_node_gemm(
    const float* __restrict__ q, const float* __restrict__ W1,
    const float* __restrict__ b1, const float* __restrict__ W2,
    const float* __restrict__ b2, float* __restrict__ xout) {
  __shared__ float hTile[16][132];  // pad 132: bank = (4*row + k) % 64, conflict-free
  const int m0 = blockIdx.x * 16;
  const int wave = threadIdx.x >> 5;
  const int lane = threadIdx.x & 31;
  const int lhalf = lane & 15;
  const bool hi = lane >= 16;

  // ---- Stage 1: h = silu(q @ W1 + b1) -------------------------------------
  {
    const int n0 = wave * 16;
    v8f acc = {};
    const float* qrow = q + (size_t)(m0 + lhalf) * NF;
    for (int k = 0; k < NF; k += 4) {
      const int ka = k + (hi ? 2 : 0);
      v2f a, b;
      // A 16x4: v0 holds K=ka (lanes 0-15: K0, lanes 16-31: K2), v1 holds K=ka+1
      a.x = qrow[ka];
      a.y = qrow[ka + 1];
      // B 4x16: row K=ka striped over lanes (N = n0 + lhalf)
      b.x = W1[(size_t)ka * NF + n0 + lhalf];
      b.y = W1[(size_t)(ka + 1) * NF + n0 + lhalf];
      acc = __builtin_amdgcn_wmma_f32_16x16x4_f32(false, a, false, b,
                                                  (short)0, acc, false, false);
    }
    const int col = n0 + lhalf;
    const float bias = b1[col];
#pragma unroll
    for (int r = 0; r < 8; ++r) {
      const int row = r + (hi ? 8 : 0);  // C/D: VGPR r -> M=r (lo) / M=r+8 (hi)
      hTile[row][col] = silu_f(acc[r] + bias);
    }
  }
  __syncthreads();

  // ---- Stage 2: x = h @ W2 + b2 -------------------------------------------
#pragma unroll
  for (int t = 0; t < 3; ++t) {
    const int n0 = (wave + 8 * t) * 16;
    v8f acc = {};
    for (int k = 0; k < NF; k += 4) {
      const int ka = k + (hi ? 2 : 0);
      v2f a, b;
      a.x = hTile[lhalf][ka];
      a.y = hTile[lhalf][ka + 1];
      b.x = W2[(size_t)ka * NF3 + n0 + lhalf];
      b.y = W2[(size_t)(ka + 1) * NF3 + n0 + lhalf];
      acc = __builtin_amdgcn_wmma_f32_16x16x4_f32(false, a, false, b,
                                                  (short)0, acc, false, false);
    }
    const int col = n0 + lhalf;
    const float bias = b2[col];
#pragma unroll
    for (int r = 0; r < 8; ++r) {
      const int row = r + (hi ? 8 : 0);
      xout[(size_t)(m0 + row) * NF3 + col] = acc[r] + bias;
    }
  }
}

// ---------------------------------------------------------------------------
// Kernel 2: seed outputs with residual:  out_q = q, out_mu = mu
// ---------------------------------------------------------------------------
__global__ void painn_init_out(const float* __restrict__ q,
                               const float* __restrict__ mu,
                               float* __restrict__ out) {
  const int i = blockIdx.x * blockDim.x + threadIdx.x;
  const int NQ = NATOMS * NF;
  const int NM = NATOMS * 3 * NF;
  if (i < NQ) {
    out[i] = q[i];
  } else if (i < NQ + NM) {
    out[i] = mu[i - NQ];
  }
}

// ---------------------------------------------------------------------------
// Kernel 3: edge phase. 128 threads per edge (one per feature), 2 edges/block.
// Wij streamed non-temporally (768 MB, touched once); x/mu gathers and the
// fp32 atomic scatter stay L2-resident (~100 MB working set << 192 MB L2).
// ---------------------------------------------------------------------------
__global__ __launch_bounds__(256) void painn_edge(
    const float* __restrict__ Wij, const float* __restrict__ dir_ij,
    const int* __restrict__ idx_i, const int* __restrict__ idx_j,
    const float* __restrict__ x, const float* __restrict__ mu,
    float* __restrict__ out_q, float* __restrict__ out_mu) {
  const int e = blockIdx.x * 2 + (threadIdx.x >> 7);
  const int f = threadIdx.x & 127;
  if (e >= NEDGES) return;

  const int ai = idx_i[e];
  const int aj = idx_j[e];

  const float* w = Wij + (size_t)e * NF3;
  const float* xj = x + (size_t)aj * NF3;
  const float* muj = mu + (size_t)aj * NF3;

  const float wq = __builtin_nontemporal_load(w + f);
  const float wR = __builtin_nontemporal_load(w + NF + f);
  const float wm = __builtin_nontemporal_load(w + 2 * NF + f);

  const float dq = wq * xj[f];
  const float dmuR = wR * xj[NF + f];
  const float dmumu = wm * xj[2 * NF + f];

  const float d0 = dir_ij[e * 3 + 0];
  const float d1 = dir_ij[e * 3 + 1];
  const float d2 = dir_ij[e * 3 + 2];

  unsafeAtomicAdd(out_q + (size_t)ai * NF + f, dq);
  unsafeAtomicAdd(out_mu + (size_t)ai * NF3 + 0 * NF + f,
                  dmuR * d0 + dmumu * muj[0 * NF + f]);
  unsafeAtomicAdd(out_mu + (size_t)ai * NF3 + 1 * NF + f,
                  dmuR * d1 + dmumu * muj[1 * NF + f]);
  unsafeAtomicAdd(out_mu + (size_t)ai * NF3 + 2 * NF + f,
                  dmuR * d2 + dmumu * muj[2 * NF + f]);
}

// ---------------------------------------------------------------------------
extern "C" void kernel_launch(void* const* d_in, const int* in_sizes, int n_in,
                              void* d_out, int out_size, void* d_ws,
                              size_t ws_size, hipStream_t stream) {
  const float* q   = (const float*)d_in[0];
  const float* mu  = (const float*)d_in[1];
  const float* Wij = (const float*)d_in[2];
  const float* dir = (const float*)d_in[3];
  const int* idx_i = (const int*)d_in[4];
  const int* idx_j = (const int*)d_in[5];
  // d_in[6] = n_atoms scalar (hardcoded as NATOMS)
  const float* W1 = (const float*)d_in[7];
  const float* b1 = (const float*)d_in[8];
  const float* W2 = (const float*)d_in[9];
  const float* b2 = (const float*)d_in[10];

  float* out_q = (float*)d_out;
  float* out_mu = out_q + (size_t)NATOMS * NF;
  float* xws = (float*)d_ws;  // [NATOMS, 384] = 30.72 MB scratch

  // 1) node MLP with fp32 WMMA
  painn_node_gemm<<<NATOMS / 16, 256, 0, stream>>>(q, W1, b1, W2, b2, xws);

  // 2) residual init of outputs
  const int total = NATOMS * NF + NATOMS * 3 * NF;
  painn_init_out<<<(total + 255) / 256, 256, 0, stream>>>(q, mu, out_q);

  // 3) edge gather/scatter
  painn_edge<<<NEDGES / 2, 256, 0, stream>>>(Wij, dir, idx_i, idx_j, xws, mu,
                                             out_q, out_mu);
}